// SecondOrderFeatureInteraction_18021682774682
// MI455X (gfx1250) — compile-verified
//
#include <hip/hip_runtime.h>

typedef float v2f __attribute__((ext_vector_type(2)));
typedef float v8f __attribute__((ext_vector_type(8)));

#define NF 32            // features
#define ED 128           // embed dim
#define RS 132           // LDS row stride (pad 4 -> bank = (4r+c)%64, conflict-free b64 reads)
#define NPAIR 496        // 32*31/2
#define WAVES_PER_BLOCK 2

__device__ __forceinline__ int triu_idx(int i, int j) {
    // row-major strict upper triangle index for N=32, k=1
    return 31 * i - ((i * (i - 1)) >> 1) + j - i - 1;
}

__global__ __launch_bounds__(WAVES_PER_BLOCK * 32)
void soi_wmma_kernel(const float* __restrict__ x, float* __restrict__ out, int n_batch) {
    __shared__ float sm[WAVES_PER_BLOCK][NF][RS];

    const int lane = threadIdx.x & 31;
    const int wave = threadIdx.x >> 5;
    const int b    = blockIdx.x * WAVES_PER_BLOCK + wave;
    if (b >= n_batch) return;   // uniform per wave: EXEC stays all-ones for live waves

    const float* __restrict__ xb = x + (size_t)b * (NF * ED);
    float (*S)[RS] = sm[wave];

    // ---- Stage batch tile (32x128 fp32 = 16KB) into LDS, fully coalesced ----
    // One global_load_b128 per lane per row: 32 lanes * 16B = one 512B row.
    #pragma unroll 8
    for (int r = 0; r < NF; ++r) {
        const float4 v = *reinterpret_cast<const float4*>(xb + r * ED + (lane << 2));
        *reinterpret_cast<float4*>(&S[r][lane << 2]) = v;
    }
    // Same-wave DS ordering is in-order; the clobber stops compiler reordering.
    asm volatile("s_wait_dscnt 0" ::: "memory");

    // ---- Gram matrix via V_WMMA_F32_16X16X4_F32 ----
    // A(16x4) and B(4x16) operand layouts coincide for Gram (B = A^T):
    // lane L holds X[base + (L&15)][k0 + 2*(L>>4) .. +1]  (v2f = 2 VGPRs).
    v8f a00 = {};   // rows 0..15  x cols 0..15
    v8f a01 = {};   // rows 0..15  x cols 16..31
    v8f a11 = {};   // rows 16..31 x cols 16..31

    const int rsel = lane & 15;
    const int coff = (lane >> 4) << 1;
    const float* __restrict__ row0 = &S[rsel][coff];
    const float* __restrict__ row1 = &S[16 + rsel][coff];

    #pragma unroll 4
    for (int k = 0; k < ED; k += 4) {
        const v2f h0 = *reinterpret_cast<const v2f*>(row0 + k);
        const v2f h1 = *reinterpret_cast<const v2f*>(row1 + k);
        // 8 args: (neg_a, A, neg_b, B, c_mod, C, reuse_a, reuse_b)
        a00 = __builtin_amdgcn_wmma_f32_16x16x4_f32(false, h0, false, h0, (short)0, a00, false, false);
        a01 = __builtin_amdgcn_wmma_f32_16x16x4_f32(false, h0, false, h1, (short)0, a01, false, false);
        a11 = __builtin_amdgcn_wmma_f32_16x16x4_f32(false, h1, false, h1, (short)0, a11, false, false);
    }

    // ---- Scatter strict upper triangle ----
    // D layout: VGPR v, lane L -> i_t = v + 8*(L>=16), j_t = L&15.
    float* __restrict__ ob = out + (size_t)b * NPAIR;
    const int j0  = lane & 15;
    const int hi8 = (lane >> 4) << 3;

    #pragma unroll
    for (int v = 0; v < 8; ++v) {            // tile (0,0): diagonal
        const int i = v + hi8, j = j0;
        if (i < j) ob[triu_idx(i, j)] = a00[v];
    }
    #pragma unroll
    for (int v = 0; v < 8; ++v) {            // tile (0,1): fully valid
        const int i = v + hi8, j = 16 + j0;
        ob[triu_idx(i, j)] = a01[v];
    }
    #pragma unroll
    for (int v = 0; v < 8; ++v) {            // tile (1,1): diagonal
        const int i = 16 + v + hi8, j = 16 + j0;
        if (i < j) ob[triu_idx(i, j)] = a11[v];
    }
}

extern "C" void kernel_launch(void* const* d_in, const int* in_sizes, int n_in,
                              void* d_out, int out_size, void* d_ws, size_t ws_size,
                              hipStream_t stream) {
    const float* x = (const float*)d_in[0];
    float* out     = (float*)d_out;
    const int n_batch = in_sizes[0] / (NF * ED);
    const int blocks  = (n_batch + WAVES_PER_BLOCK - 1) / WAVES_PER_BLOCK;
    soi_wmma_kernel<<<blocks, WAVES_PER_BLOCK * 32, 0, stream>>>(x, out, n_batch);
}